// PiNetTorch_57913339019802
// MI455X (gfx1250) — compile-verified
//
#include <hip/hip_runtime.h>

#define N_ATOMS 30000
#define N_PAIRS 600000
#define N_STRUCT 64
#define NDEPTH 3
#define FDIM 16
#define BDIM 10
#define RCUT 5.0f
#define GAMMA 3.0f

typedef __attribute__((ext_vector_type(2))) float v2f;
typedef __attribute__((ext_vector_type(8))) float v8f;

// D = A(16x4, f32) * B(4x16, f32) + C(16x16, f32)  -- exact f32 on the matrix pipe.
// Layout (ISA 7.12.2): lane l: m/n = l&15, kbase = 2*(l>>4);
//   A: {A[m][kb], A[m][kb+1]}  B: {B[kb][n], B[kb+1][n]}  C/D: vgpr r -> row r+8*(l>>4), col n.
__device__ __forceinline__ v8f wmma4(v2f a, v2f b, v8f c) {
  return __builtin_amdgcn_wmma_f32_16x16x4_f32(false, a, false, b, (short)0, c, false, false);
}

// Half-wave swap (lane <-> lane^16) via ds_swizzle_b32 SWAPX16 (xor=0x10, and=0x1f).
__device__ __forceinline__ float xor16(float v) {
#if __has_builtin(__builtin_amdgcn_ds_swizzle)
  return __int_as_float(__builtin_amdgcn_ds_swizzle(__float_as_int(v), 0x401F));
#else
  return __shfl_xor(v, 16, 32);
#endif
}

// Native CDNA5 transcendentals (v_tanh_f32 / v_exp_f32 / v_cos_f32).
__device__ __forceinline__ float fast_tanh(float x) {
#if __has_builtin(__builtin_amdgcn_tanhf)
  return __builtin_amdgcn_tanhf(x);
#else
  return tanhf(x);
#endif
}
__device__ __forceinline__ float fast_exp2(float x) {
#if __has_builtin(__builtin_amdgcn_exp2f)
  return __builtin_amdgcn_exp2f(x);
#else
  return exp2f(x);
#endif
}
__device__ __forceinline__ float cos_pi(float x) {  // cos(pi*x), x in [0,1]
#if __has_builtin(__builtin_amdgcn_cosf)
  return __builtin_amdgcn_cosf(0.5f * x);          // v_cos_f32: cos(2*pi*s)
#else
  return cosf(3.14159265358979f * x);
#endif
}

#define FULLK(k) (((k) < 8) ? fl[(k)] : fh[(k)-8])

// ---------------------------------------------------------------- init
__global__ void k_init(const float* __restrict__ prop, float* __restrict__ prop_cur,
                       float* __restrict__ out_acc, float* __restrict__ out) {
  int i = blockIdx.x * blockDim.x + threadIdx.x;
  if (i < N_ATOMS * FDIM) prop_cur[i] = prop[i];
  if (i < N_ATOMS) out_acc[i] = 0.f;
  if (i < N_STRUCT) out[i] = 0.f;
}

// ------------------------------------------- p = tanh(tanh(prop@W1+b1)@W2+b2)
__global__ void k_pp(const float* __restrict__ propc, const float* __restrict__ w1,
                     const float* __restrict__ b1, const float* __restrict__ w2,
                     const float* __restrict__ b2, float* __restrict__ pout) {
  __shared__ float sW1[256], sW2[256], sB1[16], sB2[16];
  const int tid = threadIdx.x;
  if (tid < 256) { sW1[tid] = w1[tid]; sW2[tid] = w2[tid]; }
  if (tid < 16) { sB1[tid] = b1[tid]; sB2[tid] = b2[tid]; }
  __syncthreads();
  const int wave = blockIdx.x * 8 + (tid >> 5);
  if (wave >= N_ATOMS / 16) return;
  const int lane = tid & 31, ln = lane & 15, hi = lane >> 4;
  const int atom = wave * 16 + ln;

  v8f acc = {0.f,0.f,0.f,0.f,0.f,0.f,0.f,0.f};
#pragma unroll
  for (int s = 0; s < 4; ++s) {
    const int kk = 4 * s + 2 * hi;
    v2f a = { sW1[kk * 16 + ln], sW1[(kk + 1) * 16 + ln] };
    v2f b = *(const v2f*)(propc + atom * FDIM + kk);
    acc = wmma4(a, b, acc);
  }
  float t1[8], fl[8], fh[8];
#pragma unroll
  for (int r = 0; r < 8; ++r) t1[r] = fast_tanh(acc[r] + sB1[r + 8 * hi]);
#pragma unroll
  for (int r = 0; r < 8; ++r) {          // rebuild full 16-row column per lane
    float o = xor16(t1[r]);
    fl[r] = hi ? o : t1[r];
    fh[r] = hi ? t1[r] : o;
  }
  v8f acc2 = {0.f,0.f,0.f,0.f,0.f,0.f,0.f,0.f};
#pragma unroll
  for (int s = 0; s < 4; ++s) {
    const int kk = 4 * s + 2 * hi;
    v2f a = { sW2[kk * 16 + ln], sW2[(kk + 1) * 16 + ln] };
    float bx = hi ? FULLK(4 * s + 2) : FULLK(4 * s);
    float by = hi ? FULLK(4 * s + 3) : FULLK(4 * s + 1);
    v2f b = { bx, by };
    acc2 = wmma4(a, b, acc2);
  }
  float ov[8];
#pragma unroll
  for (int r = 0; r < 8; ++r) ov[r] = fast_tanh(acc2[r] + sB2[r + 8 * hi]);
  float* dst = pout + atom * FDIM + 8 * hi;
  *(float4*)(dst)     = make_float4(ov[0], ov[1], ov[2], ov[3]);
  *(float4*)(dst + 4) = make_float4(ov[4], ov[5], ov[6], ov[7]);
}

// ----------------- per 16-pair tile: gather -> 16x160 GEMM -> basis contract
// ----------------- -> two 16x16 GEMMs -> atomic scatter into prop
__global__ void k_pair(const float* __restrict__ p, const float* __restrict__ dist,
                       const int* __restrict__ ind2, const float* __restrict__ piw,
                       const float* __restrict__ pib, const float* __restrict__ w1,
                       const float* __restrict__ w2, float* __restrict__ propc) {
  __shared__ float sPW[32 * 160];
  __shared__ float sPB[160];
  __shared__ float sW1[256], sW2[256];
  const int tid = threadIdx.x;
  for (int i = tid; i < 32 * 160; i += 256) sPW[i] = piw[i];
  if (tid < 160) sPB[tid] = pib[tid];
  if (tid < 256) { sW1[tid] = w1[tid]; sW2[tid] = w2[tid]; }
  __syncthreads();
  const int wave = blockIdx.x * 8 + (tid >> 5);
  if (wave >= N_PAIRS / 16) return;
  const int lane = tid & 31, ln = lane & 15, hi = lane >> 4;
  const int pr = wave * 16 + ln;
  const int ii = ind2[2 * pr], jj = ind2[2 * pr + 1];
  const float d = dist[pr];

  float bas[BDIM];
  {
    float x = fminf(fmaxf(d * (1.0f / RCUT), 0.f), 1.f);
    float fc = 0.5f * (cos_pi(x) + 1.0f);
    fc = (d > 0.f) ? fc : 0.f;
    const float nglog2e = -GAMMA * 1.44269504088896f;   // exp(-g t^2) = 2^(t^2 * nglog2e)
#pragma unroll
    for (int b = 0; b < BDIM; ++b) {
      float t = d - (RCUT / (BDIM - 1)) * (float)b;
      bas[b] = fc * fast_exp2(t * t * nglog2e);
    }
  }

  v2f bop[8];                           // B = pairT (32 x 16), 8 k-steps
#pragma unroll
  for (int s = 0; s < 8; ++s) {
    const int kk = 4 * s + 2 * hi;
    const float* src = (s < 4) ? (p + ii * FDIM + kk) : (p + jj * FDIM + (kk - 16));
    bop[s] = *(const v2f*)src;
  }

  float part[FDIM];
#pragma unroll
  for (int c = 0; c < FDIM; ++c) part[c] = 0.f;

#pragma unroll
  for (int t = 0; t < 10; ++t) {        // 160 output channels, 10 tiles
    v8f acc = {0.f,0.f,0.f,0.f,0.f,0.f,0.f,0.f};
#pragma unroll
    for (int s = 0; s < 8; ++s) {
      const int kk = 4 * s + 2 * hi;
      v2f a = { sPW[kk * 160 + t * 16 + ln], sPW[(kk + 1) * 160 + t * 16 + ln] };
      acc = wmma4(a, bop[s], acc);
    }
#pragma unroll
    for (int r = 0; r < 8; ++r) {
      float h = fast_tanh(acc[r] + sPB[t * 16 + r + 8 * hi]);
      const int ch0 = t * 16 + r, ch1 = ch0 + 8;     // compile-time decode
      float v0 = h * bas[ch0 % BDIM];
      float v1 = h * bas[ch1 % BDIM];
      part[ch0 / BDIM] += hi ? 0.f : v0;
      part[ch1 / BDIM] += hi ? v1 : 0.f;
    }
  }
  float inter[16];
#pragma unroll
  for (int c = 0; c < 16; ++c) inter[c] = part[c] + xor16(part[c]);

  v8f a1 = {0.f,0.f,0.f,0.f,0.f,0.f,0.f,0.f};
#pragma unroll
  for (int s = 0; s < 4; ++s) {
    const int kk = 4 * s + 2 * hi;
    v2f a = { sW1[kk * 16 + ln], sW1[(kk + 1) * 16 + ln] };
    float bx = hi ? inter[4 * s + 2] : inter[4 * s];
    float by = hi ? inter[4 * s + 3] : inter[4 * s + 1];
    v2f b = { bx, by };
    a1 = wmma4(a, b, a1);
  }
  float t1[8], fl[8], fh[8];
#pragma unroll
  for (int r = 0; r < 8; ++r) t1[r] = fast_tanh(a1[r]);
#pragma unroll
  for (int r = 0; r < 8; ++r) {
    float o = xor16(t1[r]);
    fl[r] = hi ? o : t1[r];
    fh[r] = hi ? t1[r] : o;
  }
  v8f a2 = {0.f,0.f,0.f,0.f,0.f,0.f,0.f,0.f};
#pragma unroll
  for (int s = 0; s < 4; ++s) {
    const int kk = 4 * s + 2 * hi;
    v2f a = { sW2[kk * 16 + ln], sW2[(kk + 1) * 16 + ln] };
    float bx = hi ? FULLK(4 * s + 2) : FULLK(4 * s);
    float by = hi ? FULLK(4 * s + 3) : FULLK(4 * s + 1);
    v2f b = { bx, by };
    a2 = wmma4(a, b, a2);
  }
#pragma unroll
  for (int r = 0; r < 8; ++r)
    atomicAdd(propc + ii * FDIM + 8 * hi + r, fast_tanh(a2[r]));
}

// ----------------- out_acc += tanh(prop@ow+ob) @ fw + fb
__global__ void k_out(const float* __restrict__ propc, const float* __restrict__ ow,
                      const float* __restrict__ ob, const float* __restrict__ fw,
                      const float* __restrict__ fb, float* __restrict__ out_acc) {
  __shared__ float sW[256], sB[16], sFW[16];
  const int tid = threadIdx.x;
  if (tid < 256) sW[tid] = ow[tid];
  if (tid < 16) { sB[tid] = ob[tid]; sFW[tid] = fw[tid]; }
  __syncthreads();
  const int wave = blockIdx.x * 8 + (tid >> 5);
  if (wave >= N_ATOMS / 16) return;
  const int lane = tid & 31, ln = lane & 15, hi = lane >> 4;
  const int atom = wave * 16 + ln;

  v8f acc = {0.f,0.f,0.f,0.f,0.f,0.f,0.f,0.f};
#pragma unroll
  for (int s = 0; s < 4; ++s) {
    const int kk = 4 * s + 2 * hi;
    v2f a = { sW[kk * 16 + ln], sW[(kk + 1) * 16 + ln] };
    v2f b = *(const v2f*)(propc + atom * FDIM + kk);
    acc = wmma4(a, b, acc);
  }
  float part = 0.f;
#pragma unroll
  for (int r = 0; r < 8; ++r)
    part += fast_tanh(acc[r] + sB[r + 8 * hi]) * sFW[r + 8 * hi];
  float tot = part + xor16(part);
  if (hi == 0) out_acc[atom] += tot + fb[0];
}

// ----------------- pooled = segment_sum(out_acc, ind_1)
__global__ void k_pool(const float* __restrict__ out_acc, const int* __restrict__ ind1,
                       float* __restrict__ out) {
  int i = blockIdx.x * blockDim.x + threadIdx.x;
  if (i < N_ATOMS) atomicAdd(out + ind1[i], out_acc[i]);
}

extern "C" void kernel_launch(void* const* d_in, const int* in_sizes, int n_in,
                              void* d_out, int out_size, void* d_ws, size_t ws_size,
                              hipStream_t stream) {
  (void)in_sizes; (void)n_in; (void)out_size; (void)ws_size;
  const float* prop   = (const float*)d_in[0];
  const float* dist   = (const float*)d_in[1];
  const float* pp_w1  = (const float*)d_in[2];
  const float* pp_b1  = (const float*)d_in[3];
  const float* pp_w2  = (const float*)d_in[4];
  const float* pp_b2  = (const float*)d_in[5];
  const float* pi_w   = (const float*)d_in[6];
  const float* pi_b   = (const float*)d_in[7];
  const float* ii_w1  = (const float*)d_in[8];
  const float* ii_w2  = (const float*)d_in[9];
  const float* out_w  = (const float*)d_in[10];
  const float* out_b  = (const float*)d_in[11];
  const float* out_fw = (const float*)d_in[12];
  const float* out_fb = (const float*)d_in[13];
  const int*   ind1   = (const int*)d_in[14];
  const int*   ind2   = (const int*)d_in[15];
  float* out = (float*)d_out;

  float* ws       = (float*)d_ws;
  float* prop_cur = ws;                      // 480000 f32
  float* pbuf     = ws + 480000;             // 480000 f32
  float* out_acc  = ws + 960000;             // 30000 f32

  k_init<<<(N_ATOMS * FDIM + 255) / 256, 256, 0, stream>>>(prop, prop_cur, out_acc, out);

  const int atomBlocks = (N_ATOMS / 16 + 7) / 8;     // 8 waves / block
  const int pairBlocks = (N_PAIRS / 16 + 7) / 8;
  for (int d = 0; d < NDEPTH; ++d) {
    k_pp<<<atomBlocks, 256, 0, stream>>>(prop_cur, pp_w1 + d * 256, pp_b1 + d * 16,
                                         pp_w2 + d * 256, pp_b2 + d * 16, pbuf);
    k_pair<<<pairBlocks, 256, 0, stream>>>(pbuf, dist, ind2, pi_w + d * 32 * 160,
                                           pi_b + d * 160, ii_w1 + d * 256,
                                           ii_w2 + d * 256, prop_cur);
    k_out<<<atomBlocks, 256, 0, stream>>>(prop_cur, out_w + d * 256, out_b + d * 16,
                                          out_fw + d * 16, out_fb + d, out_acc);
  }
  k_pool<<<(N_ATOMS + 255) / 256, 256, 0, stream>>>(out_acc, ind1, out);
}